// Rasterizer_68865505624179
// MI455X (gfx1250) — compile-verified
//
#include <hip/hip_runtime.h>

typedef float v2f __attribute__((ext_vector_type(2)));
typedef float v4f __attribute__((ext_vector_type(4)));
typedef float v8f __attribute__((ext_vector_type(8)));

#define M_TOTAL 512
// exp(-d2/(2*0.01^2)) == exp2(-(d2 * 5000/ln2)) ; fold sqrt(5000/ln2) into coords
#define SCALE 84.93218f   // sqrt(5000 / ln 2) = sqrt(7213.4752044448)

__global__ __launch_bounds__(256) void
Rasterizer_68865505624179_kernel(const float* __restrict__ cp, float* __restrict__ out) {
    __shared__ __align__(16) float xsS[M_TOTAL];   // pre-scaled sample x
    __shared__ __align__(16) float ysS[M_TOTAL];   // pre-scaled sample y
    __shared__ __align__(16) float spS[M_TOTAL];   // speeds

    const int b   = blockIdx.x >> 7;   // 16 batches
    const int row = blockIdx.x & 127;  // mesh row i
    const int tid = threadIdx.x;

    // ---- Stage samples + speeds for this batch into LDS (2 per thread) ----
    const float* cpb = cp + b * 128;   // (16 curves, 4 pts, 2)
    for (int k = tid; k < M_TOTAL; k += 256) {
        const int curve = k >> 5, step = k & 31;
        const float* p = cpb + curve * 8;
        const float P0x = p[0], P0y = p[1], P1x = p[2], P1y = p[3];
        const float P2x = p[4], P2y = p[5], P3x = p[6], P3y = p[7];
        const float t = (float)step * (1.0f / 31.0f);
        const float u = 1.0f - t;
        const float u2 = u * u, t2 = t * t;
        const float c0 = u2 * u, c1 = 3.0f * t * u2, c2 = 3.0f * u * t2, c3 = t2 * t;
        // NOTE: reference uses P2 for BOTH the t^2 and t^3 terms (quirk preserved)
        xsS[k] = (c0 * P0x + c1 * P1x + (c2 + c3) * P2x) * SCALE;
        ysS[k] = (c0 * P0y + c1 * P1y + (c2 + c3) * P2y) * SCALE;
        const float dxt = 3.0f * u2 * (P1x - P0x) + 6.0f * t * u * (P2x - P1x) + 3.0f * t2 * (P3x - P2x);
        const float dyt = 3.0f * u2 * (P1y - P0y) + 6.0f * t * u * (P2y - P1y) + 3.0f * t2 * (P3y - P2y);
        spS[k] = sqrtf(dxt * dxt + dyt * dyt);
    }
    __syncthreads();

    // ---- WMMA splat accumulation: 16 pixels (rows) x K=4 samples per step ----
    const int lane  = tid & 31;
    const int wave  = tid >> 5;        // 8 waves -> 8 column tiles of 16
    const int jbase = wave * 16;
    const int col   = lane & 15;       // pixel (A-matrix row) handled by this lane
    const int half  = lane >> 4;       // lanes 0-15: K=0,1 ; lanes 16-31: K=2,3

    // mesh: xm = -0.25 + (2.5/128)*j ; ym = -0.4 + (2.2/128)*(127-i)  (scaled)
    const float px = (-0.25f + 0.01953125f * (float)(jbase + col)) * SCALE;
    const float py = (-0.4f  + 0.0171875f  * (float)(127 - row))   * SCALE;

    v8f acc = {0.f, 0.f, 0.f, 0.f, 0.f, 0.f, 0.f, 0.f};

    #pragma unroll 8
    for (int kb = 0; kb < M_TOTAL; kb += 4) {
        const int k0 = kb + 2 * half;                 // even -> 8B aligned LDS pairs
        const v2f xs2 = *(const v2f*)&xsS[k0];
        const v2f ys2 = *(const v2f*)&ysS[k0];
        const v2f sp2 = *(const v2f*)&spS[k0];

        const float dx0 = xs2.x - px, dy0 = ys2.x - py;
        const float dx1 = xs2.y - px, dy1 = ys2.y - py;
        // explicit fma shape -> packs as v_pk_mul + v_pk_fma (no re-pack movs)
        const float d0 = fmaf(dy0, dy0, dx0 * dx0);
        const float d1 = fmaf(dy1, dy1, dx1 * dx1);

        v2f a;
        a.x = __builtin_amdgcn_exp2f(-d0);  // neg folds into v_exp src modifier
        a.y = __builtin_amdgcn_exp2f(-d1);
        // B = speeds replicated across all 16 columns (same per-lane K mapping as A)
        const v2f bb = sp2;

        acc = __builtin_amdgcn_wmma_f32_16x16x4_f32(
            /*neg_a=*/false, a, /*neg_b=*/false, bb,
            /*c_mod=*/(short)0, acc, /*reuse_a=*/false, /*reuse_b=*/false);
    }

    // D layout: lanes 0-15 hold rows 0..7 (N=lane), lanes 16-31 hold rows 8..15.
    // All columns equal -> lane 0 stores pixels 0..7, lane 16 stores 8..15.
    if (col == 0) {
        float* o = out + ((b * 128 + row) * 128 + jbase) + half * 8;
        v4f lo = { acc[0], acc[1], acc[2], acc[3] };
        v4f hi = { acc[4], acc[5], acc[6], acc[7] };
        *(v4f*)(o)     = lo;
        *(v4f*)(o + 4) = hi;
    }
}

extern "C" void kernel_launch(void* const* d_in, const int* in_sizes, int n_in,
                              void* d_out, int out_size, void* d_ws, size_t ws_size,
                              hipStream_t stream) {
    const float* cp = (const float*)d_in[0];  // (16,16,4,2) f32 = 2048
    float* out = (float*)d_out;               // (16,128,128) f32 = 262144
    // 2048 blocks = 16 batches x 128 rows; 256 threads = 8 wave32 column tiles
    Rasterizer_68865505624179_kernel<<<dim3(2048), dim3(256), 0, stream>>>(cp, out);
}